// Autoencoder_86526411145548
// MI455X (gfx1250) — compile-verified
//
#include <hip/hip_runtime.h>

typedef float v2f __attribute__((ext_vector_type(2)));
typedef float v4f __attribute__((ext_vector_type(4)));
typedef float v8f __attribute__((ext_vector_type(8)));

#define NRAYS 8192
#define KSEL  4096
#define LOG2E 1.4426950408889634f

// ---------------------------------------------------------------------------
// Workspace layout (floats):
//   G2   [KSEL][16]  gaussian feature vectors, pre-scaled by -0.5*log2(e)
//   glab [KSEL]      gathered labels (+-1)
//   F    [NRAYS][16] ray feature vectors
// ---------------------------------------------------------------------------

// Per-gaussian precompute: gather, Sigma = L L^T, invert via triangular
// inverse (Sigma^{-1} = L^{-T} L^{-1}), build 16-dim weight vector.
__global__ void build_gauss(const float* __restrict__ emb,
                            const float* __restrict__ chol,
                            const float* __restrict__ labels,
                            const int*   __restrict__ idx,
                            float* __restrict__ G2,
                            float* __restrict__ glab) {
    int k = blockIdx.x * blockDim.x + threadIdx.x;
    if (k >= KSEL) return;
    int id = idx[k];

    float mu[4];
#pragma unroll
    for (int i = 0; i < 4; ++i) mu[i] = emb[id * 4 + i];

    float L[4][4];
#pragma unroll
    for (int i = 0; i < 4; ++i)
#pragma unroll
        for (int j = 0; j < 4; ++j) L[i][j] = chol[id * 16 + i * 4 + j];

    // Li = L^{-1}, lower triangular (forward substitution; diag > 0).
    float Li[4][4];
    Li[0][1] = Li[0][2] = Li[0][3] = Li[1][2] = Li[1][3] = Li[2][3] = 0.0f;
    Li[0][0] = 1.0f / L[0][0];
    Li[1][1] = 1.0f / L[1][1];
    Li[2][2] = 1.0f / L[2][2];
    Li[3][3] = 1.0f / L[3][3];
    Li[1][0] = -(L[1][0] * Li[0][0]) * Li[1][1];
    Li[2][1] = -(L[2][1] * Li[1][1]) * Li[2][2];
    Li[2][0] = -(L[2][0] * Li[0][0] + L[2][1] * Li[1][0]) * Li[2][2];
    Li[3][2] = -(L[3][2] * Li[2][2]) * Li[3][3];
    Li[3][1] = -(L[3][1] * Li[1][1] + L[3][2] * Li[2][1]) * Li[3][3];
    Li[3][0] = -(L[3][0] * Li[0][0] + L[3][1] * Li[1][0] + L[3][2] * Li[2][0]) * Li[3][3];

    // A = Sigma^{-1} = Li^T Li (symmetric)
    float A[4][4];
#pragma unroll
    for (int i = 0; i < 4; ++i) {
#pragma unroll
        for (int j = i; j < 4; ++j) {
            float s = 0.0f;
#pragma unroll
            for (int m = 0; m < 4; ++m)
                if (m >= j) s += Li[m][i] * Li[m][j];
            A[i][j] = s;
            A[j][i] = s;
        }
    }

    float b[4];
#pragma unroll
    for (int i = 0; i < 4; ++i)
        b[i] = A[i][0] * mu[0] + A[i][1] * mu[1] + A[i][2] * mu[2] + A[i][3] * mu[3];
    float cq = mu[0] * b[0] + mu[1] * b[1] + mu[2] * b[2] + mu[3] * b[3];

    const float s = -0.5f * LOG2E;   // fold -0.5*q and exp->exp2 conversion
    float g[16];
    g[0]  = A[0][0] * s;  g[1]  = A[1][1] * s;  g[2]  = A[2][2] * s;  g[3]  = A[3][3] * s;
    g[4]  = 2.0f * A[0][1] * s;  g[5]  = 2.0f * A[0][2] * s;  g[6]  = 2.0f * A[0][3] * s;
    g[7]  = 2.0f * A[1][2] * s;  g[8]  = 2.0f * A[1][3] * s;  g[9]  = 2.0f * A[2][3] * s;
    g[10] = -2.0f * b[0] * s;  g[11] = -2.0f * b[1] * s;
    g[12] = -2.0f * b[2] * s;  g[13] = -2.0f * b[3] * s;
    g[14] = cq * s;  g[15] = 0.0f;

    float* dst = G2 + k * 16;
    *(v4f*)(dst + 0)  = *(v4f*)(g + 0);
    *(v4f*)(dst + 4)  = *(v4f*)(g + 4);
    *(v4f*)(dst + 8)  = *(v4f*)(g + 8);
    *(v4f*)(dst + 12) = *(v4f*)(g + 12);
    glab[k] = labels[id];
}

// Per-ray feature vector.
__global__ void build_rays(const float* __restrict__ orig,
                           const float* __restrict__ dir,
                           float* __restrict__ F) {
    int n = blockIdx.x * blockDim.x + threadIdx.x;
    if (n >= NRAYS) return;
    float p0 = orig[n * 2 + 0], p1 = orig[n * 2 + 1];
    float p2 = dir[n * 2 + 0],  p3 = dir[n * 2 + 1];
    float f[16] = { p0 * p0, p1 * p1, p2 * p2, p3 * p3,
                    p0 * p1, p0 * p2, p0 * p3, p1 * p2, p1 * p3, p2 * p3,
                    p0, p1, p2, p3, 1.0f, 0.0f };
    float* dst = F + n * 16;
    *(v4f*)(dst + 0)  = *(v4f*)(f + 0);
    *(v4f*)(dst + 4)  = *(v4f*)(f + 4);
    *(v4f*)(dst + 8)  = *(v4f*)(f + 8);
    *(v4f*)(dst + 12) = *(v4f*)(f + 12);
}

// Main fused kernel: [16 x 16] q-tiles via chained V_WMMA_F32_16X16X4_F32,
// epilogue lab*exp2(q'), row-sum reduction, sigmoid.
// Block = 256 threads (8 waves); each block owns one 16-ray tile; each wave
// handles KSEL/8 = 512 gaussians (32 column tiles).
__global__ __launch_bounds__(256) void pairwise(const float* __restrict__ F,
                                                const float* __restrict__ G2,
                                                const float* __restrict__ glab,
                                                float* __restrict__ out) {
    __shared__ float red[8][16];
    const int tid  = threadIdx.x;
    const int wave = tid >> 5;
    const int lane = tid & 31;
    const int hi   = lane >> 4;   // half-wave select: K offset +2, rows +8 in C
    const int ln   = lane & 15;
    const int rowt = blockIdx.x;

    // A fragments (ray features), fixed for the whole block's K loop.
    // Fragment layout: lane l, VGPR j -> F[row = l%16][4t + j + 2*(l>=16)]
    const float* Fr = F + (rowt * 16 + ln) * 16 + 2 * hi;
    v2f a0 = *(const v2f*)(Fr + 0);
    v2f a1 = *(const v2f*)(Fr + 4);
    v2f a2 = *(const v2f*)(Fr + 8);
    v2f a3 = *(const v2f*)(Fr + 12);

    float acc[8];
#pragma unroll
    for (int r = 0; r < 8; ++r) acc[r] = 0.0f;

    const int col0 = wave * (KSEL / 8) + ln;
    for (int ct = 0; ct < (KSEL / 8) / 16; ++ct) {
        const int col = col0 + ct * 16;
        // B fragments: lane l, VGPR j -> G[4t + j + 2*(l>=16)][col = l%16 of tile]
        const float* Gc = G2 + col * 16 + 2 * hi;
        v2f b0 = *(const v2f*)(Gc + 0);
        v2f b1 = *(const v2f*)(Gc + 4);
        v2f b2 = *(const v2f*)(Gc + 8);
        v2f b3 = *(const v2f*)(Gc + 12);

        v8f c = {0.0f, 0.0f, 0.0f, 0.0f, 0.0f, 0.0f, 0.0f, 0.0f};
        c = __builtin_amdgcn_wmma_f32_16x16x4_f32(false, a0, false, b0, (short)0, c, false, false);
        c = __builtin_amdgcn_wmma_f32_16x16x4_f32(false, a1, false, b1, (short)0, c, false, false);
        c = __builtin_amdgcn_wmma_f32_16x16x4_f32(false, a2, false, b2, (short)0, c, false, false);
        c = __builtin_amdgcn_wmma_f32_16x16x4_f32(false, a3, false, b3, (short)0, c, false, false);

        const float lab = glab[col];
#pragma unroll
        for (int r = 0; r < 8; ++r)
            acc[r] += lab * __builtin_amdgcn_exp2f(c[r]);
    }

    // Reduce across the 16 lanes of each half-wave (C layout: lane%16 = column).
#pragma unroll
    for (int m = 1; m < 16; m <<= 1) {
#pragma unroll
        for (int r = 0; r < 8; ++r)
            acc[r] += __shfl_xor(acc[r], m, 32);
    }
    // lane 0 -> rows 0..7, lane 16 -> rows 8..15
    if (ln == 0) {
#pragma unroll
        for (int r = 0; r < 8; ++r) red[wave][hi * 8 + r] = acc[r];
    }
    __syncthreads();

    if (tid < 16) {
        float s = 0.0f;
#pragma unroll
        for (int w = 0; w < 8; ++w) s += red[w][tid];
        out[rowt * 16 + tid] = 1.0f / (1.0f + __builtin_amdgcn_exp2f(-s * LOG2E));
    }
}

extern "C" void kernel_launch(void* const* d_in, const int* in_sizes, int n_in,
                              void* d_out, int out_size, void* d_ws, size_t ws_size,
                              hipStream_t stream) {
    (void)in_sizes; (void)n_in; (void)out_size; (void)ws_size;
    const float* origins    = (const float*)d_in[0];   // [N,2]
    const float* directions = (const float*)d_in[1];   // [N,2]
    const float* embeddings = (const float*)d_in[2];   // [M,4]
    const float* chol       = (const float*)d_in[3];   // [M,4,4]
    const float* labels     = (const float*)d_in[4];   // [M]
    const int*   idx        = (const int*)d_in[5];     // [K]
    float* out = (float*)d_out;                        // [N,1]

    float* ws   = (float*)d_ws;
    float* G2   = ws;                         // 16*KSEL floats (256 KB)
    float* glab = ws + 16 * KSEL;             // KSEL floats   (16 KB)
    float* F    = ws + 16 * KSEL + KSEL;      // 16*NRAYS floats (512 KB), 16B aligned

    build_gauss<<<KSEL / 256, 256, 0, stream>>>(embeddings, chol, labels, idx, G2, glab);
    build_rays<<<NRAYS / 256, 256, 0, stream>>>(origins, directions, F);
    pairwise<<<NRAYS / 16, 256, 0, stream>>>(F, G2, glab, out);
}